// YOLO4Loss_10754598109996
// MI455X (gfx1250) — compile-verified
//
#include <hip/hip_runtime.h>
#include <math.h>

// ---------------------------------------------------------------------------
// YOLOv4 loss for MI455X (gfx1250, wave32).
//   input  : (16, 255, 76, 76) f32   targets : (16, 20, 5) f32
//   out    : [loss, num_pos]
// Kernel 1: per-target prep (best anchor / gi / gj / cls) -> ws records
// Kernel 2: fused per-cell pass; async global->LDS target staging;
//           WMMA_F32_16X16X4_F32 wave reductions; atomicAdd into ws[0..3]
// Kernel 3: finalize two scalars.
// ---------------------------------------------------------------------------

#define YW   76
#define YH   76
#define YHW  5776            // 76*76
#define YNA  3
#define YNC  80
#define YCH  85              // 5+NC
#define YBS  16
#define YNGT 20
#define YCPB (YNA * YHW)     // 17328 cells per batch
#define YBPB 68              // ceil(17328/256) blocks per batch
#define YEPS 1e-7f
#define YPI  3.14159265358979323846f

typedef float v2f __attribute__((ext_vector_type(2)));
typedef float v8f __attribute__((ext_vector_type(8)));

#if defined(__gfx1250__) && __has_builtin(__builtin_amdgcn_global_load_async_to_lds_b32)
#define YOLO_ASYNC_LDS 1
typedef __attribute__((address_space(1))) int* yolo_gptr;
typedef __attribute__((address_space(3))) int* yolo_lptr;
#endif
#if defined(__gfx1250__) && __has_builtin(__builtin_amdgcn_wmma_f32_16x16x4_f32)
#define YOLO_WMMA_RED 1
#endif

// Wave32 full-lane sum via V_WMMA_F32_16X16X4_F32 reduction trick.
// A: vgpr0 = v (K=0 lanes 0-15 / K=2 lanes 16-31), vgpr1 = 0 (K=1,3).
// B: all ones.  D[m,n] = v[m] + v[m+16].
// Per-lane sum of the 8 D VGPRs = sum of rows 0-7 (lanes<16) or 8-15 (>=16);
// one shfl_xor(16) add gives the full 32-lane total on every lane.
__device__ __forceinline__ float yolo_wave_sum(float v) {
#ifdef YOLO_WMMA_RED
    v2f a; a[0] = v; a[1] = 0.0f;
    v2f b1; b1[0] = 1.0f; b1[1] = 1.0f;
    v8f c = {0.f, 0.f, 0.f, 0.f, 0.f, 0.f, 0.f, 0.f};
    c = __builtin_amdgcn_wmma_f32_16x16x4_f32(false, a, false, b1,
                                              (short)0, c, false, false);
    float s = c[0] + c[1] + c[2] + c[3] + c[4] + c[5] + c[6] + c[7];
    s += __shfl_xor(s, 16, 32);
    return s;
#else
    for (int off = 16; off > 0; off >>= 1) v += __shfl_xor(v, off, 32);
    return v;
#endif
}

__device__ __forceinline__ float yolo_sigmoid(float x) {
    return 1.0f / (1.0f + expf(-x));
}

// ---------------------------------------------------------------------------
// Kernel 1: per-target preprocessing + zero the 4 loss accumulators.
// ws layout: ws[0..3] = {conf, cls, loc, npos} accumulators (float)
//            ws[16 + (b*20+t)*8 + {0..7}] = {cx, cy, w, h, best, gi, gj, cls}
//            (last 4 fields stored as int bit patterns)
// ---------------------------------------------------------------------------
__global__ __launch_bounds__(320) void yolo_prep(const float* __restrict__ targets,
                                                 float* __restrict__ ws) {
    const int tid = threadIdx.x;
    if (tid < 4) ws[tid] = 0.0f;
    if (tid >= YBS * YNGT) return;

    const float* tg = targets + (size_t)tid * 5;
    const float cx = tg[0] * (float)YW;
    const float cy = tg[1] * (float)YH;
    const float gw = tg[2] * (float)YW;
    const float gh = tg[3] * (float)YH;
    const int   cls = (int)tg[4];

    int gi = (int)floorf(cx); gi = min(max(gi, 0), YW - 1);
    int gj = (int)floorf(cy); gj = min(max(gj, 0), YH - 1);

    // scaled anchors = ANCHORS / 8
    const float AW[3] = {1.5f, 2.375f, 5.0f};
    const float AH[3] = {2.0f, 4.5f, 3.5f};
    float bestr = -1.0f; int best = 0;
#pragma unroll
    for (int a = 0; a < YNA; ++a) {
        const float inter = fminf(gw, AW[a]) * fminf(gh, AH[a]);
        const float uni   = gw * gh + AW[a] * AH[a] - inter;
        const float r     = inter / uni;
        if (r > bestr) { bestr = r; best = a; }   // first max wins (argmax)
    }

    float* rec = ws + 16 + (size_t)tid * 8;
    rec[0] = cx; rec[1] = cy; rec[2] = gw; rec[3] = gh;
    int* ri = (int*)rec;
    ri[4] = best; ri[5] = gi; ri[6] = gj; ri[7] = cls;
}

// ---------------------------------------------------------------------------
// Kernel 2: fused per-cell loss pass. One thread per (b, a, j, i) cell.
// ---------------------------------------------------------------------------
__global__ __launch_bounds__(256) void yolo_main(const float* __restrict__ inp,
                                                 float* __restrict__ ws) {
    __shared__ float s_rec[YNGT * 8];   // 640 B: this batch's target records

    const int b     = blockIdx.x / YBPB;
    const int chunk = blockIdx.x % YBPB;
    const int local = chunk * 256 + threadIdx.x;

    const float* rec_base = ws + 16 + (size_t)b * YNGT * 8;

    // ---- stage target records into LDS (async DMA path on CDNA5) ----
    if (threadIdx.x < YNGT * 8) {
#ifdef YOLO_ASYNC_LDS
        __builtin_amdgcn_global_load_async_to_lds_b32(
            (yolo_gptr)(rec_base + threadIdx.x),
            (yolo_lptr)(&s_rec[threadIdx.x]),
            0, 0);
#else
        s_rec[threadIdx.x] = rec_base[threadIdx.x];
#endif
    }
#ifdef YOLO_ASYNC_LDS
#if __has_builtin(__builtin_amdgcn_s_wait_asynccnt)
    __builtin_amdgcn_s_wait_asynccnt(0);
#endif
#endif
    __syncthreads();

    float acc_conf = 0.0f, acc_cls = 0.0f, acc_loc = 0.0f, acc_np = 0.0f;

    if (local < YCPB) {
        const int a   = local / YHW;
        const int rem = local - a * YHW;
        const int j   = rem / YW;
        const int i   = rem - j * YW;

        const float AW[3] = {1.5f, 2.375f, 5.0f};
        const float AH[3] = {2.0f, 4.5f, 3.5f};

        // channel-major layout: cp[ch * YHW]
        const float* cp = inp + ((size_t)b * (YNA * YCH) + (size_t)a * YCH) * YHW + rem;
        const float x    = cp[0];
        const float y    = cp[(size_t)1 * YHW];
        const float tw   = cp[(size_t)2 * YHW];
        const float th   = cp[(size_t)3 * YHW];
        const float conf = yolo_sigmoid(cp[(size_t)4 * YHW]);

        // NB: reference adds the ROW index to x and the COL index to y.
        const float pbx = (float)j + x;
        const float pby = (float)i + y;
        const float pbw = expf(tw) * AW[a];
        const float pbh = expf(th) * AH[a];

        const float pminx = pbx - 0.5f * pbw, pmaxx = pbx + 0.5f * pbw;
        const float pminy = pby - 0.5f * pbh, pmaxy = pby + 0.5f * pbh;
        const float areap = pbw * pbh;

        float max_iou = 0.0f;
        int   pos = -1;
#pragma unroll 4
        for (int t = 0; t < YNGT; ++t) {
            const float* r = &s_rec[t * 8];
            const float gx = r[0], gy = r[1], gw = r[2], gh = r[3];
            const float gminx = gx - 0.5f * gw, gmaxx = gx + 0.5f * gw;
            const float gminy = gy - 0.5f * gh, gmaxy = gy + 0.5f * gh;
            const float iw = fmaxf(fminf(pmaxx, gmaxx) - fmaxf(pminx, gminx), 0.0f);
            const float ih = fmaxf(fminf(pmaxy, gmaxy) - fmaxf(pminy, gminy), 0.0f);
            const float inter = iw * ih;
            const float uni   = areap + gw * gh - inter;
            max_iou = fmaxf(max_iou, inter / fmaxf(uni, 1e-6f));
            const int* ri = (const int*)r;
            if (ri[4] == a && ri[6] == j && ri[5] == i) pos = t;  // last write wins
        }

        if (pos >= 0) {
            // conf BCE vs target 1
            const float p = fminf(fmaxf(conf, YEPS), 1.0f - YEPS);
            acc_conf = -logf(p);
            acc_np   = 1.0f;

            // CIoU localization loss vs matched target box
            const float* r = &s_rec[pos * 8];
            const float gx = r[0], gy = r[1], gw = r[2], gh = r[3];
            const float gminx = gx - 0.5f * gw, gmaxx = gx + 0.5f * gw;
            const float gminy = gy - 0.5f * gh, gmaxy = gy + 0.5f * gh;
            const float iw = fmaxf(fminf(pmaxx, gmaxx) - fmaxf(pminx, gminx), 0.0f);
            const float ih = fmaxf(fminf(pmaxy, gmaxy) - fmaxf(pminy, gminy), 0.0f);
            const float inter = iw * ih;
            const float uni   = areap + gw * gh - inter;
            const float iou   = inter / fmaxf(uni, 1e-6f);
            const float cd2   = (pbx - gx) * (pbx - gx) + (pby - gy) * (pby - gy);
            const float encw  = fmaxf(fmaxf(pmaxx, gmaxx) - fminf(pminx, gminx), 0.0f);
            const float ench  = fmaxf(fmaxf(pmaxy, gmaxy) - fminf(pminy, gminy), 0.0f);
            const float encd  = encw * encw + ench * ench;
            const float ciou  = iou - cd2 / fmaxf(encd, 1e-6f);
            const float dv    = atanf(pbw / fmaxf(pbh, 1e-6f)) -
                                atanf(gw  / fmaxf(gh,  1e-6f));
            const float vv    = (4.0f / (YPI * YPI)) * dv * dv;
            const float alpha = vv / fmaxf(1.0f - iou + vv, 1e-6f);
            acc_loc = 1.0f - ciou + alpha * vv;

            // class BCE with label smoothing (only positive cells touch cls ch.)
            const int cls = ((const int*)r)[7];
            const float t_on  = 1.0f - 0.01f + 0.01f / (float)YNC;   // 0.990125
            const float t_off = 0.01f / (float)YNC;                  // 0.000125
            float s = 0.0f;
            for (int c = 0; c < YNC; ++c) {
                float pc = yolo_sigmoid(cp[(size_t)(5 + c) * YHW]);
                pc = fminf(fmaxf(pc, YEPS), 1.0f - YEPS);
                const float tsm = (c == cls) ? t_on : t_off;
                s += -tsm * logf(pc) - (1.0f - tsm) * logf(1.0f - pc);
            }
            acc_cls = s;
        } else if (max_iou <= 0.5f) {
            // no-object conf BCE vs target 0 (ignored if IoU > 0.5)
            const float p = fminf(fmaxf(conf, YEPS), 1.0f - YEPS);
            acc_conf = -logf(1.0f - p);
        }
    }

    // ---- block reduction: WMMA wave sums (EXEC is all-1s here), LDS combine ----
    const float wc = yolo_wave_sum(acc_conf);
    const float wk = yolo_wave_sum(acc_cls);
    const float wl = yolo_wave_sum(acc_loc);
    const float wn = yolo_wave_sum(acc_np);

    __shared__ float s_red[8][4];
    const int wave = threadIdx.x >> 5;
    const int lane = threadIdx.x & 31;
    if (lane == 0) {
        s_red[wave][0] = wc; s_red[wave][1] = wk;
        s_red[wave][2] = wl; s_red[wave][3] = wn;
    }
    __syncthreads();
    if (threadIdx.x < 4) {
        float s = 0.0f;
#pragma unroll
        for (int w2 = 0; w2 < 8; ++w2) s += s_red[w2][threadIdx.x];
        atomicAdd(&ws[threadIdx.x], s);
    }
}

// ---------------------------------------------------------------------------
// Kernel 3: finalize.
// ---------------------------------------------------------------------------
__global__ void yolo_final(const float* __restrict__ ws, float* __restrict__ out) {
    if (threadIdx.x == 0) {
        out[0] = ws[0] + ws[1] + ws[2];      // conf + cls + loc (lambdas = 1)
        out[1] = fmaxf(ws[3], 1.0f);         // num_pos
    }
}

extern "C" void kernel_launch(void* const* d_in, const int* in_sizes, int n_in,
                              void* d_out, int out_size, void* d_ws, size_t ws_size,
                              hipStream_t stream) {
    (void)in_sizes; (void)n_in; (void)out_size; (void)ws_size;
    const float* inp = (const float*)d_in[0];
    const float* tgt = (const float*)d_in[1];
    float* out = (float*)d_out;
    float* wsf = (float*)d_ws;   // needs 64 B + 320*8*4 B = ~10.3 KB

    yolo_prep<<<1, 320, 0, stream>>>(tgt, wsf);
    yolo_main<<<YBS * YBPB, 256, 0, stream>>>(inp, wsf);
    yolo_final<<<1, 32, 0, stream>>>(wsf, out);
}